// GraphNeuralNetwork_70437463655122
// MI455X (gfx1250) — compile-verified
//
#include <hip/hip_runtime.h>
#include <hip/hip_bf16.h>

typedef __attribute__((ext_vector_type(2))) float v2f;
typedef __attribute__((ext_vector_type(8))) float v8f;

#define HEADS 4
#define OUT 128
#define HO 512
#define NEG_SLOPE 0.2f
#define LN_EPS 1e-5f

// ---------------------------------------------------------------------------
// fp32 WMMA GEMM:  Y[M,N] = X[M,K] @ W[K,N] + bias[N]
// One wave per 16x64 output strip (4 accumulators) so each A fragment feeds
// 4 V_WMMA_F32_16X16X4_F32 ops; K-loop steps by 4.
// A layout: lanes 0-15 hold row M=lane%16 (K=k0,k0+1); lanes 16-31 K=k0+2,+3.
// B layout: lanes 0-15 hold col N=lane%16 (K=k0,k0+1); lanes 16-31 K=k0+2,+3.
// C/D layout: vgpr r -> row r + 8*(lane>=16), col lane%16.
// ---------------------------------------------------------------------------
__global__ void gemm_bias_wmma(const float* __restrict__ X,
                               const float* __restrict__ W,
                               const float* __restrict__ bias,
                               float* __restrict__ Y,
                               int M, int K, int N)
{
    int wave = (int)((blockIdx.x * blockDim.x + threadIdx.x) >> 5);
    int lane = (int)(threadIdx.x & 31);
    int stripsN = N >> 6;                       // 64-wide strips
    int tileM = wave / stripsN;
    int strip = wave - tileM * stripsN;
    if (tileM * 16 >= M) return;                // wave-uniform

    int half = lane >> 4;                       // 0 or 1
    int l16  = lane & 15;
    int rowA  = tileM * 16 + l16;
    int colB0 = strip * 64 + l16;

    v8f acc0 = {0.f,0.f,0.f,0.f,0.f,0.f,0.f,0.f};
    v8f acc1 = acc0, acc2 = acc0, acc3 = acc0;
    const float* xrow = X + (size_t)rowA * K;

    for (int k0 = 0; k0 < K; k0 += 4) {
        int ka = k0 + half * 2;
        v2f a;
        a.x = xrow[ka];
        a.y = xrow[ka + 1];
        const float* w0 = W + (size_t)ka * N + colB0;
        const float* w1 = W + (size_t)(ka + 1) * N + colB0;
        v2f b0, b1, b2, b3;
        b0.x = w0[0];  b0.y = w1[0];
        b1.x = w0[16]; b1.y = w1[16];
        b2.x = w0[32]; b2.y = w1[32];
        b3.x = w0[48]; b3.y = w1[48];
        acc0 = __builtin_amdgcn_wmma_f32_16x16x4_f32(false, a, false, b0, (short)0, acc0, false, false);
        acc1 = __builtin_amdgcn_wmma_f32_16x16x4_f32(false, a, false, b1, (short)0, acc1, false, false);
        acc2 = __builtin_amdgcn_wmma_f32_16x16x4_f32(false, a, false, b2, (short)0, acc2, false, false);
        acc3 = __builtin_amdgcn_wmma_f32_16x16x4_f32(false, a, false, b3, (short)0, acc3, false, false);
    }

    int rbase = tileM * 16 + half * 8;
    float bv0 = bias[colB0];
    float bv1 = bias[colB0 + 16];
    float bv2 = bias[colB0 + 32];
    float bv3 = bias[colB0 + 48];
#pragma unroll
    for (int r = 0; r < 8; ++r) {
        float* yr = Y + (size_t)(rbase + r) * N + colB0;
        yr[0]  = acc0[r] + bv0;
        yr[16] = acc1[r] + bv1;
        yr[32] = acc2[r] + bv2;
        yr[48] = acc3[r] + bv3;
    }
}

// ---------------------------------------------------------------------------
// utility: fill buffer with constant
// ---------------------------------------------------------------------------
__global__ void fill_f32(float* __restrict__ p, float v, long n)
{
    long i = (long)blockIdx.x * blockDim.x + threadIdx.x;
    if (i < n) p[i] = v;
}

__device__ __forceinline__ float lrelu(float x)
{
    return (x >= 0.f) ? x : NEG_SLOPE * x;
}

// float atomic max via integer punning (valid for mixed signs with -FLT_MAX init)
__device__ __forceinline__ void atomicMaxF(float* addr, float val)
{
    if (val >= 0.f) atomicMax((int*)addr, __float_as_int(val));
    else            atomicMin((unsigned int*)addr, __float_as_uint(val));
}

// ---------------------------------------------------------------------------
// per (edge, head): logit = sum_c att[h,c] * leaky_relu(xl[src,h,c]+xr[dst,h,c])
// then segment-max into m[dst,h]
// ---------------------------------------------------------------------------
__global__ void edge_logits(const float* __restrict__ xl,
                            const float* __restrict__ xr,
                            const int* __restrict__ ei, int E, int Etot,
                            const float* __restrict__ att,
                            float* __restrict__ logits,
                            float* __restrict__ mmax)
{
    int idx = (int)(blockIdx.x * blockDim.x + threadIdx.x);
    if (idx >= Etot * HEADS) return;
    int e = idx >> 2, h = idx & 3;
    int s, d;
    if (e < E) { s = ei[e]; d = ei[E + e]; }
    else       { s = d = e - E; }

    const float* pl = xl + (size_t)s * HO + h * OUT;
    const float* pr = xr + (size_t)d * HO + h * OUT;
    const float* pa = att + h * OUT;

    float sum = 0.f;
#pragma unroll 4
    for (int c = 0; c < OUT; c += 4) {
        float4 a = *(const float4*)(pl + c);
        float4 b = *(const float4*)(pr + c);
        float4 w = *(const float4*)(pa + c);
        sum += w.x * lrelu(a.x + b.x);
        sum += w.y * lrelu(a.y + b.y);
        sum += w.z * lrelu(a.z + b.z);
        sum += w.w * lrelu(a.w + b.w);
    }
    logits[idx] = sum;
    atomicMaxF(&mmax[d * HEADS + h], sum);
}

// ---------------------------------------------------------------------------
// per (edge, head): ex = exp(logit - m[dst]); denom[dst] += ex (in-place store)
// ---------------------------------------------------------------------------
__global__ void edge_exp(const int* __restrict__ ei, int E, int Etot,
                         float* __restrict__ logits,
                         const float* __restrict__ mmax,
                         float* __restrict__ denom)
{
    int idx = (int)(blockIdx.x * blockDim.x + threadIdx.x);
    if (idx >= Etot * HEADS) return;
    int e = idx >> 2, h = idx & 3;
    int d = (e < E) ? ei[E + e] : (e - E);
    float ex = expf(logits[idx] - mmax[d * HEADS + h]);
    logits[idx] = ex;
    atomicAdd(&denom[d * HEADS + h], ex);
}

// ---------------------------------------------------------------------------
// per (edge, 4 channels): accum[dst,h,c] += alpha * xl[src,h,c]
// ---------------------------------------------------------------------------
__global__ void edge_scatter(const float* __restrict__ xl,
                             const int* __restrict__ ei, int E, int Etot,
                             const float* __restrict__ ex,
                             const float* __restrict__ denom,
                             float* __restrict__ accum)
{
    long idx = (long)blockIdx.x * blockDim.x + threadIdx.x;
    if (idx >= (long)Etot * (HO / 4)) return;
    int e = (int)(idx >> 7);          // HO/4 = 128 groups per edge
    int j = (int)(idx & 127);         // channel group, 4 channels each
    int s, d;
    if (e < E) { s = ei[e]; d = ei[E + e]; }
    else       { s = d = e - E; }
    int h = j >> 5;                   // (j*4)/128
    float alpha = ex[e * HEADS + h] / (denom[d * HEADS + h] + 1e-16f);
    float4 v = *(const float4*)(xl + (size_t)s * HO + j * 4);
    float* o = accum + (size_t)d * HO + j * 4;
    atomicAdd(o + 0, alpha * v.x);
    atomicAdd(o + 1, alpha * v.y);
    atomicAdd(o + 2, alpha * v.z);
    atomicAdd(o + 3, alpha * v.w);
}

// ---------------------------------------------------------------------------
// per node (128 threads): y = mean_h(accum) + bias -> LayerNorm -> ReLU
// ---------------------------------------------------------------------------
__global__ void finalize_ln(const float* __restrict__ accum,
                            const float* __restrict__ bias,
                            const float* __restrict__ g,
                            const float* __restrict__ be,
                            float* __restrict__ xout)
{
    int node = blockIdx.x;
    int c = threadIdx.x;               // 0..127
    const float* a = accum + (size_t)node * HO;
    float y = 0.25f * (a[c] + a[OUT + c] + a[2 * OUT + c] + a[3 * OUT + c]) + bias[c];

    __shared__ float red[OUT];
    red[c] = y;
    __syncthreads();
#pragma unroll
    for (int s = OUT / 2; s > 0; s >>= 1) {
        if (c < s) red[c] += red[c + s];
        __syncthreads();
    }
    float mu = red[0] * (1.0f / OUT);
    __syncthreads();

    float dd = y - mu;
    red[c] = dd * dd;
    __syncthreads();
#pragma unroll
    for (int s = OUT / 2; s > 0; s >>= 1) {
        if (c < s) red[c] += red[c + s];
        __syncthreads();
    }
    float var = red[0] * (1.0f / OUT);
    float r = rsqrtf(var + LN_EPS);
    float o = dd * r * g[c] + be[c];
    xout[(size_t)node * OUT + c] = fmaxf(o, 0.f);
}

// ---------------------------------------------------------------------------
extern "C" void kernel_launch(void* const* d_in, const int* in_sizes, int n_in,
                              void* d_out, int out_size, void* d_ws, size_t ws_size,
                              hipStream_t stream)
{
    const float* nodef = (const float*)d_in[0];
    const int*   ei    = (const int*)d_in[1];
    const float* Wl0   = (const float*)d_in[2];
    const float* bl0   = (const float*)d_in[3];
    const float* Wr0   = (const float*)d_in[4];
    const float* br0   = (const float*)d_in[5];
    const float* att0  = (const float*)d_in[6];
    const float* b0    = (const float*)d_in[7];
    const float* g0    = (const float*)d_in[8];
    const float* be0   = (const float*)d_in[9];
    const float* Wl    = (const float*)d_in[10];
    const float* bl    = (const float*)d_in[11];
    const float* Wr    = (const float*)d_in[12];
    const float* br    = (const float*)d_in[13];
    const float* att   = (const float*)d_in[14];
    const float* bb    = (const float*)d_in[15];
    const float* gg    = (const float*)d_in[16];
    const float* bbe   = (const float*)d_in[17];

    int E = in_sizes[1] / 2;                 // 131072
    int F = in_sizes[2] / HO;                // 8
    int numNodes = in_sizes[0] / F;          // 8192
    int Etot = E + numNodes;                 // self loops appended

    size_t nHO = (size_t)numNodes * HO;
    float* xl    = (float*)d_ws;
    float* xr    = xl + nHO;
    float* accum = xr + nHO;
    float* exb   = accum + nHO;
    float* m     = exb + (size_t)Etot * HEADS;
    float* denom = m + (size_t)numNodes * HEADS;
    float* xcur  = denom + (size_t)numNodes * HEADS;
    size_t needed = (size_t)(xcur + (size_t)numNodes * OUT - (float*)d_ws) * sizeof(float);
    if (ws_size < needed) return;

    const int BLK = 256;
    int gemmBlocks    = (int)((((size_t)(numNodes / 16) * (HO / 64)) * 32 + BLK - 1) / BLK);
    int ehBlocks      = (Etot * HEADS + BLK - 1) / BLK;
    int scatterBlocks = (int)(((long)Etot * (HO / 4) + BLK - 1) / BLK);
    int fillHeads     = (numNodes * HEADS + BLK - 1) / BLK;
    int fillAcc       = (int)((nHO + BLK - 1) / BLK);

    for (int layer = 0; layer < 3; ++layer) {
        const float *X, *pWl, *pbl, *pWr, *pbr, *patt, *pb, *pg, *pbe;
        int K;
        if (layer == 0) {
            X = nodef; K = F;
            pWl = Wl0; pbl = bl0; pWr = Wr0; pbr = br0;
            patt = att0; pb = b0; pg = g0; pbe = be0;
        } else {
            int i = layer - 1;
            X = xcur; K = OUT;
            pWl = Wl + (size_t)i * OUT * HO; pbl = bl + (size_t)i * HO;
            pWr = Wr + (size_t)i * OUT * HO; pbr = br + (size_t)i * HO;
            patt = att + (size_t)i * HEADS * OUT;
            pb = bb + (size_t)i * OUT; pg = gg + (size_t)i * OUT; pbe = bbe + (size_t)i * OUT;
        }

        gemm_bias_wmma<<<gemmBlocks, BLK, 0, stream>>>(X, pWl, pbl, xl, numNodes, K, HO);
        gemm_bias_wmma<<<gemmBlocks, BLK, 0, stream>>>(X, pWr, pbr, xr, numNodes, K, HO);

        fill_f32<<<fillHeads, BLK, 0, stream>>>(m, -3.402823466e38f, (long)numNodes * HEADS);
        fill_f32<<<fillHeads, BLK, 0, stream>>>(denom, 0.f, (long)numNodes * HEADS);
        fill_f32<<<fillAcc, BLK, 0, stream>>>(accum, 0.f, (long)nHO);

        edge_logits<<<ehBlocks, BLK, 0, stream>>>(xl, xr, ei, E, Etot, patt, exb, m);
        edge_exp<<<ehBlocks, BLK, 0, stream>>>(ei, E, Etot, exb, m, denom);
        edge_scatter<<<scatterBlocks, BLK, 0, stream>>>(xl, ei, E, Etot, exb, denom, accum);

        float* xo = (layer == 2) ? (float*)d_out : xcur;
        finalize_ln<<<numNodes, OUT, 0, stream>>>(accum, pb, pg, pbe, xo);
    }
}